// GCN_14766097563851
// MI455X (gfx1250) — compile-verified
//
#include <hip/hip_runtime.h>
#include <math.h>

// f32 WMMA fragment types: A(16x4)=2 VGPR, B(4x16)=2 VGPR, C/D(16x16)=8 VGPR
typedef __attribute__((ext_vector_type(2))) float v2f;
typedef __attribute__((ext_vector_type(8))) float v8f;

// ---------------------------------------------------------------- utilities
__global__ void k_zero(float* __restrict__ p, long long n) {
  long long i = (long long)blockIdx.x * blockDim.x + threadIdx.x;
  if (i < n) p[i] = 0.f;
}

// deg[dst] += 1 per edge (integer counts in f32: exact, order-independent)
__global__ void k_degree(const int* __restrict__ ei, float* __restrict__ deg, int E) {
  int e = blockIdx.x * blockDim.x + threadIdx.x;
  if (e < E) atomicAdd(&deg[ei[E + e]], 1.0f);
}

// dinv = rsqrt(deg + 1)   (self loop)
__global__ void k_dinv(float* __restrict__ deg, int N) {
  int n = blockIdx.x * blockDim.x + threadIdx.x;
  if (n < N) deg[n] = rsqrtf(deg[n] + 1.0f);
}

// ------------------------------------------------- attention-gate pipeline
// one wave per node: s = dot(x[n], wg) + bg ; store exp(s); block partial Z
__global__ void k_gate(const float* __restrict__ X, const float* __restrict__ wg,
                       const float* __restrict__ bgate, int layer,
                       float* __restrict__ sexp, float* __restrict__ zpart, int N) {
  int w = threadIdx.x >> 5;
  int n = blockIdx.x * 8 + w;
  int lane = threadIdx.x & 31;
  float e = 0.f;
  if (n < N) {
    const float* xr = X + (size_t)n * 64;
    float p = xr[2 * lane] * wg[2 * lane] + xr[2 * lane + 1] * wg[2 * lane + 1];
#pragma unroll
    for (int off = 16; off > 0; off >>= 1) p += __shfl_xor(p, off, 32);
    e = __expf(p + bgate[layer]);
    if (lane == 0) sexp[n] = e;
  }
  __shared__ float sm[8];
  if (lane == 0) sm[w] = e;
  __syncthreads();
  if (threadIdx.x == 0) {
    float z = 0.f;
#pragma unroll
    for (int i = 0; i < 8; ++i) z += sm[i];
    zpart[blockIdx.x] = z;
  }
}

// deterministic single-block tree reduction of partials -> *Z
__global__ void k_reduce(const float* __restrict__ part, int n, float* __restrict__ Z) {
  __shared__ float sm[256];
  float s = 0.f;
  for (int i = threadIdx.x; i < n; i += 256) s += part[i];
  sm[threadIdx.x] = s;
  __syncthreads();
  for (int off = 128; off > 0; off >>= 1) {
    if (threadIdx.x < off) sm[threadIdx.x] += sm[threadIdx.x + off];
    __syncthreads();
  }
  if (threadIdx.x == 0) *Z = sm[0];
}

// block-partial weighted column sums: ppart[b][d] = sum_n exp(s[n]) * x[n][d]
__global__ void k_pool_partial(const float* __restrict__ X, const float* __restrict__ sexp,
                               float* __restrict__ ppart, int N) {
  int d = threadIdx.x & 63;
  int g = threadIdx.x >> 6;  // 0..3
  float acc = 0.f;
  for (int n = blockIdx.x * 4 + g; n < N; n += gridDim.x * 4)
    acc += sexp[n] * X[(size_t)n * 64 + d];
  __shared__ float sm[256];
  sm[threadIdx.x] = acc;
  __syncthreads();
  if (g == 0)
    ppart[(size_t)blockIdx.x * 64 + d] = sm[d] + sm[64 + d] + sm[128 + d] + sm[192 + d];
}

// single block: pooled = reduce(ppart)/Z ; hsum += pooled @ Wpred[layer] + bpred[layer]
__global__ void k_pool_final(const float* __restrict__ ppart, int nblocks,
                             const float* __restrict__ Z, const float* __restrict__ Wpred,
                             const float* __restrict__ bpred, int layer,
                             float* __restrict__ hsum) {
  __shared__ float pooled[64];
  int d = threadIdx.x;  // blockDim == 64
  float acc = 0.f;
  for (int b = 0; b < nblocks; ++b) acc += ppart[(size_t)b * 64 + d];
  pooled[d] = acc / (*Z);
  __syncthreads();
  if (d < 32) {
    const float* Wp = Wpred + (size_t)layer * 64 * 32;
    float h = bpred[layer * 32 + d];
#pragma unroll 8
    for (int k = 0; k < 64; ++k) h += pooled[k] * Wp[k * 32 + d];
    hsum[d] += h;  // single block: deterministic accumulation across layers
  }
}

// ------------------------------------------------------------- WMMA GEMM
// H[N,64] = X[N,64] @ W[64,64] via V_WMMA_F32_16X16X4_F32.
// One wave -> one 16-row strip: 4 N-tiles x 16 K-steps of WMMA(K=4).
__global__ void k_gemm_wmma(const float* __restrict__ X, const float* __restrict__ W,
                            float* __restrict__ Hout, int N) {
  int wave = blockIdx.x * (blockDim.x >> 5) + (threadIdx.x >> 5);
  int lane = threadIdx.x & 31;
  int m0 = wave * 16;
  if (m0 >= N) return;              // whole wave exits together: EXEC stays all-ones
  int half = lane >> 4;             // 0: K,K+1 ; 1: K+2,K+3
  int l15 = lane & 15;

  // A fragments for all 16 K-steps (ISA 16x4 f32 layout)
  v2f a[16];
  const float* xrow = X + (size_t)(m0 + l15) * 64;
#pragma unroll
  for (int kt = 0; kt < 16; ++kt) {
    int k = kt * 4 + half * 2;
    a[kt].x = xrow[k];
    a[kt].y = xrow[k + 1];
  }

#pragma unroll
  for (int nt = 0; nt < 4; ++nt) {
    v8f acc = {0.f, 0.f, 0.f, 0.f, 0.f, 0.f, 0.f, 0.f};
#pragma unroll
    for (int kt = 0; kt < 16; ++kt) {
      int kbase = kt * 4 + half * 2;
      v2f b;
      b.x = W[(size_t)(kbase)     * 64 + nt * 16 + l15];
      b.y = W[(size_t)(kbase + 1) * 64 + nt * 16 + l15];
      acc = __builtin_amdgcn_wmma_f32_16x16x4_f32(false, a[kt], false, b,
                                                  (short)0, acc, false, false);
    }
    // C/D layout: VGPR r -> row m0 + r + half*8, col nt*16 + l15
#pragma unroll
    for (int r = 0; r < 8; ++r)
      Hout[(size_t)(m0 + r + half * 8) * 64 + nt * 16 + l15] = acc[r];
  }
}

// acc = h * dinv^2 + bconv[layer]   (self loop + bias folded into acc init)
__global__ void k_selfloop_bias(const float* __restrict__ H, const float* __restrict__ dinv,
                                const float* __restrict__ bconv, int layer,
                                float* __restrict__ acc, int N) {
  long long idx = (long long)blockIdx.x * blockDim.x + threadIdx.x;
  if (idx >= (long long)N * 64) return;
  int n = (int)(idx >> 6), d = (int)(idx & 63);
  float di = dinv[n];
  acc[idx] = H[idx] * di * di + bconv[layer * 64 + d];
}

// edge scatter: acc[dst] += h[src] * dinv[src]*dinv[dst]  (one thread per edge-column)
__global__ void k_scatter(const float* __restrict__ H, const int* __restrict__ ei,
                          const float* __restrict__ dinv, float* __restrict__ acc, int E) {
  long long idx = (long long)blockIdx.x * blockDim.x + threadIdx.x;
  if (idx >= (long long)E * 64) return;
  int e = (int)(idx >> 6);
  int d = (int)(idx & 63);
  int s = ei[e], t = ei[E + e];
  float coef = dinv[s] * dinv[t];
  atomicAdd(&acc[(size_t)t * 64 + d], H[(size_t)s * 64 + d] * coef);
}

// out[0] = exp(h0) * exp((hsum/4) . beta)
__global__ void k_final(const float* __restrict__ hsum, const float* __restrict__ beta,
                        const float* __restrict__ h0, float* __restrict__ out) {
  int o = threadIdx.x;  // 32 threads
  float v = hsum[o] * 0.25f * beta[o];
#pragma unroll
  for (int off = 16; off > 0; off >>= 1) v += __shfl_down(v, off, 32);
  if (o == 0) out[0] = __expf(h0[0]) * __expf(v);
}

// ---------------------------------------------------------------- launcher
extern "C" void kernel_launch(void* const* d_in, const int* in_sizes, int n_in,
                              void* d_out, int out_size, void* d_ws, size_t ws_size,
                              hipStream_t stream) {
  (void)n_in; (void)out_size; (void)ws_size;
  const float* x     = (const float*)d_in[0];
  const int*   ei    = (const int*)d_in[1];   // [2,E] row-major: src then dst
  const float* Wconv = (const float*)d_in[2]; // [3,64,64]
  const float* bconv = (const float*)d_in[3]; // [3,64]
  const float* Wpred = (const float*)d_in[4]; // [4,64,32]
  const float* bpred = (const float*)d_in[5]; // [4,32]
  const float* Wgate = (const float*)d_in[6]; // [4,64]
  const float* bgate = (const float*)d_in[7]; // [4]
  const float* h0    = (const float*)d_in[8]; // [1]
  const float* beta  = (const float*)d_in[9]; // [32]

  const int N = in_sizes[0] / 64;
  const int E = in_sizes[1] / 2;
  const long long NH = (long long)N * 64;

  // workspace layout (floats): 3 feature buffers + dinv + sexp + partials + scalars
  float* buf0  = (float*)d_ws;
  float* buf1  = buf0 + NH;           // h = x @ Wconv
  float* buf2  = buf1 + NH;
  float* dinv  = buf2 + NH;           // N  (deg -> dinv in place)
  float* sexp  = dinv + N;            // N
  const int nZ = (N + 7) / 8;
  float* zpart = sexp + N;            // nZ
  const int PBLK = 240;
  float* ppart = zpart + nZ;          // PBLK*64
  float* hsum  = ppart + (size_t)PBLK * 64;  // 32
  float* Zsc   = hsum + 32;           // 1

  dim3 b256(256);
  k_zero<<<(unsigned)((N + 255) / 256), b256, 0, stream>>>(dinv, N);
  k_zero<<<1, 64, 0, stream>>>(hsum, 33);  // hsum + Zsc
  k_degree<<<(unsigned)((E + 255) / 256), b256, 0, stream>>>(ei, dinv, E);
  k_dinv<<<(unsigned)((N + 255) / 256), b256, 0, stream>>>(dinv, N);

  const float* cur = x;
  float* accbuf[3] = {buf0, buf2, buf0};
  const unsigned gemmGrid = (unsigned)(((N + 15) / 16 + 7) / 8);
  const long long scatTot = (long long)E * 64;

  for (int i = 0; i < 3; ++i) {
    // attention pooling head on pre-conv features
    k_gate<<<(unsigned)nZ, b256, 0, stream>>>(cur, Wgate + i * 64, bgate, i, sexp, zpart, N);
    k_reduce<<<1, b256, 0, stream>>>(zpart, nZ, Zsc);
    k_pool_partial<<<PBLK, b256, 0, stream>>>(cur, sexp, ppart, N);
    k_pool_final<<<1, 64, 0, stream>>>(ppart, PBLK, Zsc, Wpred, bpred, i, hsum);
    // GCN conv
    float* acc = accbuf[i];
    k_gemm_wmma<<<gemmGrid, b256, 0, stream>>>(cur, Wconv + (size_t)i * 64 * 64, buf1, N);
    k_selfloop_bias<<<(unsigned)((NH + 255) / 256), b256, 0, stream>>>(buf1, dinv, bconv, i, acc, N);
    k_scatter<<<(unsigned)((scatTot + 255) / 256), b256, 0, stream>>>(buf1, ei, dinv, acc, E);
    cur = acc;
  }

  // final pooling (layer 3) on post-conv features
  k_gate<<<(unsigned)nZ, b256, 0, stream>>>(cur, Wgate + 3 * 64, bgate, 3, sexp, zpart, N);
  k_reduce<<<1, b256, 0, stream>>>(zpart, nZ, Zsc);
  k_pool_partial<<<PBLK, b256, 0, stream>>>(cur, sexp, ppart, N);
  k_pool_final<<<1, 64, 0, stream>>>(ppart, PBLK, Zsc, Wpred, bpred, 3, hsum);

  k_final<<<1, 32, 0, stream>>>(hsum, beta, h0, (float*)d_out);
}